// EchoStateMemory_18949395710457
// MI455X (gfx1250) — compile-verified
//
#include <hip/hip_runtime.h>
#include <hip/hip_bf16.h>
#include <math.h>

typedef float v2f __attribute__((ext_vector_type(2)));
typedef float v8f __attribute__((ext_vector_type(8)));

#define B_   2
#define L_   512
#define LM1  511
#define D_   1024
#define DH   512
#define NRES 4096
#define MROWS (B_ * LM1)     // 1022
#define SCAN_BLOCKS 128      // 32768 threads: 4-way K-split per reservoir row

// ---------------------------------------------------------------------------
// C[M x N] = A[M x K] * W[N x K]^T  using V_WMMA_F32_16X16X4_F32.
// One wave per 16x16 output tile; 8 waves (256 thr) per block.
// A fragment (16x4 f32): lane m = lane&15, VGPR pair holds K = {0,1} (lanes 0-15)
// or K = {2,3} (lanes 16-31). B fragment mirrors with lane = N column.
// projRemap=1 maps logical row m (b*511+t) onto x rows (b*512+t).
// ---------------------------------------------------------------------------
__global__ void gemm_wmma_f32(const float* __restrict__ A,
                              const float* __restrict__ W,
                              float* __restrict__ C,
                              int M, int N, int K, int projRemap) {
  const int lane  = threadIdx.x & 31;
  const int wave  = threadIdx.x >> 5;
  const int tileN = blockIdx.x * 8 + wave;
  const int tileM = blockIdx.y;
  if (tileN * 16 >= N) return;
  const int half = lane >> 4;     // 0: K{0,1}/M rows 0-7 ; 1: K{2,3}/M rows 8-15
  const int l16  = lane & 15;
  int mrow = tileM * 16 + l16;
  if (mrow > M - 1) mrow = M - 1;                 // clamp padded rows
  int arow = mrow;
  if (projRemap) arow += (mrow >= LM1) ? 1 : 0;   // skip last token of batch 0
  const int ncol = tileN * 16 + l16;
  const float* Ap = A + (size_t)arow * K + (half << 1);
  const float* Wp = W + (size_t)ncol * K + (half << 1);
  v8f acc = {};
  for (int k = 0; k < K; k += 4) {
    v2f a = *(const v2f*)(Ap + k);
    v2f b = *(const v2f*)(Wp + k);
    acc = __builtin_amdgcn_wmma_f32_16x16x4_f32(false, a, false, b,
                                                (short)0, acc, false, false);
  }
#pragma unroll
  for (int r = 0; r < 8; ++r) {
    int row = tileM * 16 + r + half * 8;
    if (row < M) C[(size_t)row * N + ncol] = acc[r];
  }
}

// ---------------------------------------------------------------------------
// Row L2-normalize (EPS=1e-6); for epi (which==1) also scale by w_t=(t+1)/512.
// ---------------------------------------------------------------------------
__global__ void norm_kernel(const float* __restrict__ ks, const float* __restrict__ ke,
                            float* __restrict__ kns, float* __restrict__ kne_w) {
  const int row = blockIdx.x, which = blockIdx.y, tid = threadIdx.x;
  const float* src = (which == 0 ? ks : ke) + (size_t)row * DH;
  float*       dst = (which == 0 ? kns : kne_w) + (size_t)row * DH;
  __shared__ float red[256];
  float v0 = src[tid], v1 = src[tid + 256];
  red[tid] = v0 * v0 + v1 * v1;
  __syncthreads();
  for (int s = 128; s > 0; s >>= 1) {
    if (tid < s) red[tid] += red[tid + s];
    __syncthreads();
  }
  float inv = 1.0f / fmaxf(sqrtf(red[0]), 1e-6f);
  if (which == 1) {
    int t = (row < LM1) ? row : row - LM1;
    inv *= (float)(t + 1) * (1.0f / (float)L_);
  }
  dst[tid] = v0 * inv;
  dst[tid + 256] = v1 * inv;
}

// ---------------------------------------------------------------------------
// Sequential EMA scan -> fire flags (independent of reservoir state).
// One block per batch. a = 0.95^(96/512).
// ---------------------------------------------------------------------------
__global__ void fire_kernel(const float* __restrict__ ks, const float* __restrict__ ke,
                            int* __restrict__ fire) {
  __shared__ float ema_s[DH], ema_e[DH];
  __shared__ float4 red[256];
  const int b = blockIdx.x, tid = threadIdx.x;
  ema_s[tid] = 0.0f; ema_s[tid + 256] = 0.0f;
  ema_e[tid] = 0.0f; ema_e[tid + 256] = 0.0f;
  __syncthreads();
  const float a = 0.99042861f;           // 0.95 ** (96/512)
  const float oma = 1.0f - a;
  for (int t = 0; t < LM1; ++t) {
    const float* kr = ks + ((size_t)b * LM1 + t) * DH;
    const float* er = ke + ((size_t)b * LM1 + t) * DH;
    float4 p = make_float4(0.f, 0.f, 0.f, 0.f);
    for (int i = tid; i < DH; i += 256) {
      float v = kr[i], w = er[i];
      float ds = v - ema_s[i], de = w - ema_e[i];
      p.x += ds * ds; p.y += v * v;
      p.z += de * de; p.w += w * w;
    }
    red[tid] = p;
    __syncthreads();
    for (int s = 128; s > 0; s >>= 1) {
      if (tid < s) {
        float4 q = red[tid + s];
        red[tid].x += q.x; red[tid].y += q.y; red[tid].z += q.z; red[tid].w += q.w;
      }
      __syncthreads();
    }
    if (tid == 0) {
      float4 r = red[0];       // err^2 >= 0.49*norm^2  <=>  err >= 0.7*norm
      fire[b * LM1 + t] = (r.x >= 0.49f * r.y) || (r.z >= 0.49f * r.w) ? 1 : 0;
    }
    for (int i = tid; i < DH; i += 256) {
      ema_s[i] = a * ema_s[i] + oma * kr[i];
      ema_e[i] = a * ema_e[i] + oma * er[i];
    }
    __syncthreads();
  }
}

// ---------------------------------------------------------------------------
// Device-wide sense-reversing barrier.
// ---------------------------------------------------------------------------
__device__ __forceinline__ void grid_sync(unsigned* cnt, unsigned* gen, unsigned nblk) {
  __syncthreads();
  if (threadIdx.x == 0) {
    __threadfence();
    volatile unsigned* vgen = (volatile unsigned*)gen;
    unsigned g = *vgen;
    if (atomicAdd(cnt, 1u) == nblk - 1u) {
      *cnt = 0u;
      __threadfence();
      *vgen = g + 1u;
    } else {
      while (*vgen == g) { __builtin_amdgcn_s_sleep(1); }
    }
  }
  __syncthreads();
}

// ---------------------------------------------------------------------------
// Persistent reservoir recurrence: 511 steps of h = fire ? tanh(W h + u) : h.
// 32768 threads: each reservoir row (8192 = 4096 x {sem,epi}) is handled by
// FOUR adjacent lanes, each covering a 1024-wide K-quarter with 2 partial
// accumulators per batch (4 independent FMA chains/thread -> latency hidden
// with 1024 waves device-wide). Quarter sums combine via __shfl_xor in-wave.
// W traffic stays minimal: each W element read exactly once per step; both
// W_res matrices (128 MB fp32) are L2-resident (192 MB L2), so steady state
// streams from L2, not HBM. h staged in LDS (32 KB/block).
// hbuf layout: [parity][res][batch][4096]; final state lands in parity 1.
// ---------------------------------------------------------------------------
__global__ void reservoir_scan(const float* __restrict__ Wsem,
                               const float* __restrict__ Wepi,
                               const float* __restrict__ usem,
                               const float* __restrict__ uepi,
                               const int* __restrict__ fire,
                               float* __restrict__ hbuf,
                               unsigned* __restrict__ bar, int nblk) {
  const int tid  = threadIdx.x;
  const int gid  = blockIdx.x * 256 + tid;
  const int kq   = gid & 3;                 // K quarter: [kq*1024, kq*1024+1024)
  const int row8 = gid >> 2;                // 0..8191
  const int res  = row8 >> 12;              // 0=sem, 1=epi (uniform per block)
  const int row  = row8 & (NRES - 1);
  const float* Wr = (res == 0 ? Wsem : Wepi) + (size_t)row * NRES + kq * 1024;
  const float* u  = (res == 0 ? usem : uepi);
  __shared__ __align__(16) float hsh[2][NRES];    // 32 KB: h for both batches
  int p = 0;
  for (int t = 0; t < LM1; ++t) {
    const float* hp = hbuf + ((size_t)p * 4 + res * 2) * NRES;
    for (int i = tid * 4; i < NRES; i += 1024) {
      *(float4*)(&hsh[0][i]) = *(const float4*)(hp + i);
      *(float4*)(&hsh[1][i]) = *(const float4*)(hp + NRES + i);
    }
    __syncthreads();
    const float* h0p = &hsh[0][kq * 1024];
    const float* h1p = &hsh[1][kq * 1024];
    float a0 = 0.f, b0 = 0.f, a1 = 0.f, b1 = 0.f;   // 4 independent chains
    for (int k = 0; k < 1024; k += 8) {
      float4 wA = *(const float4*)(Wr + k);
      float4 wB = *(const float4*)(Wr + k + 4);
      float4 xA0 = *(const float4*)(h0p + k);
      float4 xB0 = *(const float4*)(h0p + k + 4);
      float4 xA1 = *(const float4*)(h1p + k);
      float4 xB1 = *(const float4*)(h1p + k + 4);
      a0 += wA.x * xA0.x + wA.y * xA0.y + wA.z * xA0.z + wA.w * xA0.w;
      b0 += wB.x * xB0.x + wB.y * xB0.y + wB.z * xB0.z + wB.w * xB0.w;
      a1 += wA.x * xA1.x + wA.y * xA1.y + wA.z * xA1.z + wA.w * xA1.w;
      b1 += wB.x * xB1.x + wB.y * xB1.y + wB.z * xB1.z + wB.w * xB1.w;
    }
    float acc0 = a0 + b0;
    float acc1 = a1 + b1;
    acc0 += __shfl_xor(acc0, 1); acc0 += __shfl_xor(acc0, 2);
    acc1 += __shfl_xor(acc1, 1); acc1 += __shfl_xor(acc1, 2);
    if (kq == 0) {
      const int f0 = fire[t], f1 = fire[LM1 + t];
      float n0 = f0 ? tanhf(acc0 + u[(size_t)t * NRES + row])         : hsh[0][row];
      float n1 = f1 ? tanhf(acc1 + u[(size_t)(LM1 + t) * NRES + row]) : hsh[1][row];
      float* hq = hbuf + ((size_t)(1 - p) * 4 + res * 2) * NRES;
      hq[row] = n0;
      hq[NRES + row] = n1;
    }
    grid_sync(bar, bar + 1, (unsigned)nblk);
    p ^= 1;
  }
}

// ---------------------------------------------------------------------------
// Per-batch scalar gate: g = sigmoid(x[b,-1,:] . w + b0)
// ---------------------------------------------------------------------------
__global__ void gate_kernel(const float* __restrict__ x, const float* __restrict__ gw,
                            const float* __restrict__ gb, float* __restrict__ g) {
  __shared__ float red[256];
  const int b = blockIdx.x, tid = threadIdx.x;
  const float* xr = x + ((size_t)b * L_ + (L_ - 1)) * D_;
  float p = 0.0f;
  for (int i = tid; i < D_; i += 256) p += xr[i] * gw[i];
  red[tid] = p;
  __syncthreads();
  for (int s = 128; s > 0; s >>= 1) {
    if (tid < s) red[tid] += red[tid + s];
    __syncthreads();
  }
  if (tid == 0) g[b] = 1.0f / (1.0f + expf(-(red[0] + gb[0])));
}

// ---------------------------------------------------------------------------
// r[b][j] = g[b] * (h_final . readout_w[j]),  j<512 sem / j>=512 epi
// hfin layout: [res][batch][4096]
// ---------------------------------------------------------------------------
__global__ void readout_kernel(const float* __restrict__ hfin,
                               const float* __restrict__ srw,
                               const float* __restrict__ erw,
                               const float* __restrict__ g, float* __restrict__ r) {
  const int gid = blockIdx.x * 256 + threadIdx.x;   // 0..2047
  const int b = gid >> 10, j = gid & (D_ - 1);
  const float *h, *w;
  if (j < DH) { h = hfin + (size_t)b * NRES;       w = srw + (size_t)j * NRES; }
  else        { h = hfin + (size_t)(2 + b) * NRES; w = erw + (size_t)(j - DH) * NRES; }
  float acc = 0.0f;
  for (int k = 0; k < NRES; k += 4) {
    float4 hv = *(const float4*)(h + k);
    float4 wv = *(const float4*)(w + k);
    acc += hv.x * wv.x + hv.y * wv.y + hv.z * wv.z + hv.w * wv.w;
  }
  r[gid] = g[b] * acc;
}

// ---------------------------------------------------------------------------
// y = r @ out_proj_w^T + bias, then LayerNorm. One block per batch.
// ---------------------------------------------------------------------------
__global__ void final_kernel(const float* __restrict__ r, const float* __restrict__ W,
                             const float* __restrict__ bias,
                             const float* __restrict__ gamma,
                             const float* __restrict__ beta, float* __restrict__ out) {
  __shared__ __align__(16) float rsh[D_];
  __shared__ float ysh[D_];
  __shared__ float red[256], red2[256];
  const int b = blockIdx.x, tid = threadIdx.x;
  for (int i = tid; i < D_; i += 256) rsh[i] = r[b * D_ + i];
  __syncthreads();
  float s = 0.0f, s2 = 0.0f;
  for (int o = 0; o < 4; ++o) {
    const int d = tid + o * 256;
    const float* Wr = W + (size_t)d * D_;
    float acc = bias[d];
    for (int k = 0; k < D_; k += 4) {
      float4 wv = *(const float4*)(Wr + k);
      acc += wv.x * rsh[k] + wv.y * rsh[k + 1] + wv.z * rsh[k + 2] + wv.w * rsh[k + 3];
    }
    ysh[d] = acc;
    s += acc;
    s2 += acc * acc;
  }
  red[tid] = s; red2[tid] = s2;
  __syncthreads();
  for (int st = 128; st > 0; st >>= 1) {
    if (tid < st) { red[tid] += red[tid + st]; red2[tid] += red2[tid + st]; }
    __syncthreads();
  }
  const float mean = red[0] * (1.0f / (float)D_);
  const float var  = red2[0] * (1.0f / (float)D_) - mean * mean;
  const float rstd = rsqrtf(var + 1e-5f);
  for (int i = tid; i < D_; i += 256) {
    out[b * D_ + i] = (ysh[i] - mean) * rstd * gamma[i] + beta[i];
  }
}

// ---------------------------------------------------------------------------
extern "C" void kernel_launch(void* const* d_in, const int* in_sizes, int n_in,
                              void* d_out, int out_size, void* d_ws, size_t ws_size,
                              hipStream_t stream) {
  const float* x        = (const float*)d_in[0];
  const float* sem_pw   = (const float*)d_in[1];
  const float* epi_pw   = (const float*)d_in[2];
  const float* sem_rw   = (const float*)d_in[3];
  const float* epi_rw   = (const float*)d_in[4];
  const float* ngw      = (const float*)d_in[5];
  const float* ngb      = (const float*)d_in[6];
  const float* opw      = (const float*)d_in[7];
  const float* opb      = (const float*)d_in[8];
  const float* gamma    = (const float*)d_in[9];
  const float* beta     = (const float*)d_in[10];
  const float* Wres_sem = (const float*)d_in[11];
  const float* Win_sem  = (const float*)d_in[12];
  const float* Wres_epi = (const float*)d_in[13];
  const float* Win_epi  = (const float*)d_in[14];
  float* out = (float*)d_out;

  // Workspace layout (256B aligned chunks)
  size_t off = 0;
  auto take = [&](size_t bytes) {
    size_t r = off;
    off += (bytes + 255) & ~(size_t)255;
    return r;
  };
  char* ws = (char*)d_ws;
  float* ks    = (float*)(ws + take((size_t)MROWS * DH * 4));
  float* ke    = (float*)(ws + take((size_t)MROWS * DH * 4));
  float* kns   = (float*)(ws + take((size_t)MROWS * DH * 4));
  float* kne_w = (float*)(ws + take((size_t)MROWS * DH * 4));
  float* usem  = (float*)(ws + take((size_t)MROWS * NRES * 4));
  float* uepi  = (float*)(ws + take((size_t)MROWS * NRES * 4));
  int*   fire  = (int*)  (ws + take((size_t)MROWS * 4));
  float* hbuf  = (float*)(ws + take((size_t)2 * 4 * NRES * 4));   // [p][res][b][4096]
  float* gbuf  = (float*)(ws + take(2 * 4));
  float* rbuf  = (float*)(ws + take((size_t)B_ * D_ * 4));
  unsigned* bar = (unsigned*)(ws + take(256));

  // Zero recurrence state + barrier counters (deterministic per launch)
  hipMemsetAsync(hbuf, 0, (size_t)2 * 4 * NRES * 4, stream);
  hipMemsetAsync(bar, 0, 256, stream);

  const dim3 blk(256);

  // 1) Projections (WMMA f32 GEMMs): ks/ke = xw @ W_proj^T
  gemm_wmma_f32<<<dim3(DH / 16 / 8, (MROWS + 15) / 16), blk, 0, stream>>>(
      x, sem_pw, ks, MROWS, DH, D_, 1);
  gemm_wmma_f32<<<dim3(DH / 16 / 8, (MROWS + 15) / 16), blk, 0, stream>>>(
      x, epi_pw, ke, MROWS, DH, D_, 1);

  // 2) Row-normalize (epi also scaled by w_t)
  norm_kernel<<<dim3(MROWS, 2), blk, 0, stream>>>(ks, ke, kns, kne_w);

  // 3) Input drives (WMMA f32 GEMMs): u = kn @ W_in^T
  gemm_wmma_f32<<<dim3(NRES / 16 / 8, (MROWS + 15) / 16), blk, 0, stream>>>(
      kns, Win_sem, usem, MROWS, NRES, DH, 0);
  gemm_wmma_f32<<<dim3(NRES / 16 / 8, (MROWS + 15) / 16), blk, 0, stream>>>(
      kne_w, Win_epi, uepi, MROWS, NRES, DH, 0);

  // 4) Fire flags (sequential EMA scan, decoupled from reservoir)
  fire_kernel<<<2, blk, 0, stream>>>(ks, ke, fire);

  // 5) Persistent reservoir recurrence with device-wide barrier
  reservoir_scan<<<SCAN_BLOCKS, blk, 0, stream>>>(
      Wres_sem, Wres_epi, usem, uepi, fire, hbuf, bar, SCAN_BLOCKS);

  // 6) Gate, readout, out-proj + LayerNorm
  gate_kernel<<<2, blk, 0, stream>>>(x, ngw, ngb, gbuf);
  readout_kernel<<<(B_ * D_) / 256, blk, 0, stream>>>(
      hbuf + (size_t)4 * NRES /* parity 1 */, sem_rw, epi_rw, gbuf, rbuf);
  final_kernel<<<2, blk, 0, stream>>>(rbuf, opw, opb, gamma, beta, out);

  (void)in_sizes; (void)n_in; (void)out_size; (void)ws_size;
}